// ZeroTPruenImportance_38706245272241
// MI455X (gfx1250) — compile-verified
//
#include <hip/hip_runtime.h>
#include <math.h>

typedef float v2f __attribute__((ext_vector_type(2)));
typedef float v8f __attribute__((ext_vector_type(8)));
typedef float f4  __attribute__((ext_vector_type(4)));

#define NN    1024
#define BB    8
#define HH    12
#define HEADS (BB * HH)   // 96

// ---------------------------------------------------------------------------
// B2 = A @ A per head, fp32, using V_WMMA_F32_16X16X4_F32.
// 128 threads = 4 wave32s; macro-tile 32x32 (each wave owns a 16x16 C tile);
// K staged through LDS in chunks of 16 (4 WMMAs per chunk per wave).
//
// Fragment layouts per CDNA5 ISA 7.12.2:
//   A 16x4 : lanes 0-15 -> M=lane, VGPR0=K0,VGPR1=K1; lanes 16-31 -> K2,K3
//   B 4x16 : lanes 0-15 -> N=lane, VGPR0=K0,VGPR1=K1; lanes 16-31 -> K2,K3
//   C 16x16: VGPR r -> M = r + 8*(lane/16), N = lane%16
// ---------------------------------------------------------------------------
__global__ __launch_bounds__(128) void sq_gemm_wmma(const float* __restrict__ A,
                                                    float* __restrict__ C) {
  const int head = blockIdx.z;
  const float* __restrict__ Ah = A + (size_t)head * NN * NN;
  float* __restrict__ Ch = C + (size_t)head * NN * NN;
  const int tm = blockIdx.y * 32;
  const int tn = blockIdx.x * 32;

  __shared__ float lA[32][17];   // 32 rows x 16 k (+1 pad)
  __shared__ float lB[16][33];   // 16 k    x 32 cols (+1 pad)

  const int t    = threadIdx.x;
  const int lane = t & 31;
  const int wave = t >> 5;
  const int wm   = (wave >> 1) * 16;   // wave's M offset within macro-tile
  const int wn   = (wave & 1)  * 16;   // wave's N offset within macro-tile
  const int lrow = lane & 15;          // M (for A) / N (for B) index
  const int lhi  = lane >> 4;          // selects K pair {0,1} vs {2,3}

  v8f acc = {};

  for (int kc = 0; kc < NN; kc += 16) {
    __syncthreads();
    { // fill lA: rows tm..tm+31, cols kc..kc+15 (float4 per thread)
      const int row = t >> 2;
      const int c   = (t & 3) * 4;
      const f4 v = *(const f4*)(Ah + (size_t)(tm + row) * NN + kc + c);
      lA[row][c + 0] = v.x; lA[row][c + 1] = v.y;
      lA[row][c + 2] = v.z; lA[row][c + 3] = v.w;
    }
    { // fill lB: rows kc..kc+15, cols tn..tn+31 (float4 per thread)
      const int row = t >> 3;
      const int c   = (t & 7) * 4;
      const f4 v = *(const f4*)(Ah + (size_t)(kc + row) * NN + tn + c);
      lB[row][c + 0] = v.x; lB[row][c + 1] = v.y;
      lB[row][c + 2] = v.z; lB[row][c + 3] = v.w;
    }
    __syncthreads();

#pragma unroll
    for (int k0 = 0; k0 < 16; k0 += 4) {
      v2f a, b;
      a.x = lA[wm + lrow][k0 + 2 * lhi + 0];
      a.y = lA[wm + lrow][k0 + 2 * lhi + 1];
      b.x = lB[k0 + 2 * lhi + 0][wn + lrow];
      b.y = lB[k0 + 2 * lhi + 1][wn + lrow];
      // (neg_a, A, neg_b, B, c_mod, C, reuse_a, reuse_b)
      acc = __builtin_amdgcn_wmma_f32_16x16x4_f32(false, a, false, b,
                                                  (short)0, acc, false, false);
    }
  }

#pragma unroll
  for (int r = 0; r < 8; ++r) {
    const int row = tm + wm + r + 8 * lhi;
    const int col = tn + wn + lrow;
    Ch[(size_t)row * NN + col] = acc[r];
  }
}

// ---------------------------------------------------------------------------
// s_out[head][n] = sum_m M[head][m][n] * s_in[head][m]
// uniform!=0 -> s_in treated as 1/N everywhere (first iteration).
// One thread per output column; block covers 256 columns (coalesced rows).
// ---------------------------------------------------------------------------
__global__ __launch_bounds__(256) void gemv_pass(const float* __restrict__ M,
                                                 const float* __restrict__ s_in,
                                                 float* __restrict__ s_out,
                                                 int uniform) {
  const int head = blockIdx.y;
  const int col  = blockIdx.x * 256 + threadIdx.x;
  const float* __restrict__ Mh = M + (size_t)head * NN * NN;

  __shared__ float sh[NN];
  if (uniform) {
    for (int i = threadIdx.x; i < NN; i += 256) sh[i] = 1.0f / (float)NN;
  } else {
    const float* __restrict__ sv = s_in + (size_t)head * NN;
    for (int i = threadIdx.x; i < NN; i += 256) sh[i] = sv[i];
  }
  __syncthreads();

  float acc = 0.0f;
#pragma unroll 8
  for (int m = 0; m < NN; ++m) {
    acc = fmaf(Mh[(size_t)m * NN + col], sh[m], acc);
  }
  s_out[(size_t)head * NN + col] = acc;
}

// ---------------------------------------------------------------------------
// importance[b][n-1] = sqrt(sum_h s[b,h,n]^2), for n = 1..N-1
// ---------------------------------------------------------------------------
__global__ __launch_bounds__(256) void finalize_norm(const float* __restrict__ s,
                                                     float* __restrict__ out) {
  const int b = blockIdx.y;
  const int n = blockIdx.x * 256 + threadIdx.x;
  if (n == 0 || n >= NN) return;
  float sum = 0.0f;
#pragma unroll
  for (int h = 0; h < HH; ++h) {
    const float v = s[((size_t)(b * HH + h)) * NN + n];
    sum = fmaf(v, v, sum);
  }
  out[(size_t)b * (NN - 1) + (n - 1)] = sqrtf(sum);
}

extern "C" void kernel_launch(void* const* d_in, const int* in_sizes, int n_in,
                              void* d_out, int out_size, void* d_ws, size_t ws_size,
                              hipStream_t stream) {
  (void)in_sizes; (void)n_in; (void)out_size;
  const float* A  = (const float*)d_in[0];
  float* out      = (float*)d_out;

  const size_t b2_bytes = (size_t)HEADS * NN * NN * sizeof(float); // 384 MB
  const size_t s_bytes  = (size_t)HEADS * NN * sizeof(float);      // 384 KB

  const dim3 gemvGrid(NN / 256, HEADS);
  const dim3 finGrid(NN / 256, BB);

  if (ws_size >= b2_bytes + 2 * s_bytes) {
    // Fast path: B2 = A^2 via WMMA, then s = s0 * (A^2)^4  (4 GEMV passes).
    float* B2 = (float*)d_ws;
    float* sa = (float*)((char*)d_ws + b2_bytes);
    float* sb = sa + (size_t)HEADS * NN;

    const dim3 gemmGrid(NN / 32, NN / 32, HEADS);
    sq_gemm_wmma<<<gemmGrid, 128, 0, stream>>>(A, B2);

    gemv_pass<<<gemvGrid, 256, 0, stream>>>(B2, nullptr, sa, 1);
    float* cur = sa; float* nxt = sb;
    for (int i = 0; i < 3; ++i) {
      gemv_pass<<<gemvGrid, 256, 0, stream>>>(B2, cur, nxt, 0);
      float* tmp = cur; cur = nxt; nxt = tmp;
    }
    finalize_norm<<<finGrid, 256, 0, stream>>>(cur, out);
  } else if (ws_size >= 2 * s_bytes) {
    // Fallback: plain 8 GEMV passes over A.
    float* sa = (float*)d_ws;
    float* sb = sa + (size_t)HEADS * NN;

    gemv_pass<<<gemvGrid, 256, 0, stream>>>(A, nullptr, sa, 1);
    float* cur = sa; float* nxt = sb;
    for (int i = 0; i < 7; ++i) {
      gemv_pass<<<gemvGrid, 256, 0, stream>>>(A, cur, nxt, 0);
      float* tmp = cur; cur = nxt; nxt = tmp;
    }
    finalize_norm<<<finGrid, 256, 0, stream>>>(cur, out);
  }
}